// GVPStub_28329604284738
// MI455X (gfx1250) — compile-verified
//
#include <hip/hip_runtime.h>

typedef __attribute__((ext_vector_type(2))) float v2f;
typedef __attribute__((ext_vector_type(8))) float v8f;

#define HDIM 64
#define FDIM 128
#define WSTRIDE 80  // 64 + 16 dword pad: half-waves hit disjoint LDS banks

#if defined(__has_builtin)
#if __has_builtin(__builtin_amdgcn_global_load_async_to_lds_b128)
#define HAVE_ASYNC_LDS 1
#endif
#if __has_builtin(__builtin_amdgcn_s_wait_asynccnt)
#define HAVE_WAIT_ASYNC 1
#endif
#endif

#if defined(HAVE_ASYNC_LDS)
typedef int v4i_ __attribute__((vector_size(16)));
typedef __attribute__((address_space(1))) v4i_ gv4i;  // global b128 chunk
typedef __attribute__((address_space(3))) v4i_ lv4i;  // LDS b128 chunk
#endif

__device__ __forceinline__ void wait_async0() {
#if defined(HAVE_WAIT_ASYNC)
  __builtin_amdgcn_s_wait_asynccnt(0);
#elif defined(HAVE_ASYNC_LDS)
  asm volatile("s_wait_asynccnt 0x0" ::: "memory");
#endif
}

__global__ void zero_f32v4(float4* __restrict__ p, int n4) {
  int i = blockIdx.x * blockDim.x + threadIdx.x;
  if (i < n4) p[i] = make_float4(0.f, 0.f, 0.f, 0.f);
}

__global__ void deg_count(const int* __restrict__ dst, float* __restrict__ deg, int E) {
  int i = blockIdx.x * blockDim.x + threadIdx.x;
  if (i < E) atomicAdd(&deg[dst[i]], 1.0f);
}

// UPDATE=false: Y[:, y_col_ofs:+64] = relu(A[:, a_col_ofs:+K] @ W + b)
// UPDATE=true : Y[:, y_col_ofs:+64] = relu(Y + A[:, a_col_ofs:+K] @ W + deg*b)
// One wave per 16x16 tile; W staged in LDS (async global->LDS when available).
template <int K, bool UPDATE>
__launch_bounds__(256)
__global__ void gemm_fused(const float* __restrict__ A, int lda, int a_col_ofs,
                           const float* __restrict__ W,
                           const float* __restrict__ bias,
                           const float* __restrict__ deg,
                           float* __restrict__ Y, int y_col_ofs,
                           int n_row_tiles) {
  __shared__ float Wlds[K * WSTRIDE];
  // Cooperative stage of W [K,64] into bank-padded LDS: K*16 b128 chunks.
  for (int idx = threadIdx.x; idx < K * 16; idx += 256) {
    const int row = idx >> 4;
    const int c4  = (idx & 15) * 4;
#if defined(HAVE_ASYNC_LDS)
    __builtin_amdgcn_global_load_async_to_lds_b128(
        (gv4i*)(W + (size_t)row * HDIM + c4),
        (lv4i*)(&Wlds[row * WSTRIDE + c4]), 0, 0);
#else
    *(float4*)(&Wlds[row * WSTRIDE + c4]) =
        *(const float4*)(W + (size_t)row * HDIM + c4);
#endif
  }
  wait_async0();
  __syncthreads();  // all waves must reach this before any tail-tile exit

  const int gw = blockIdx.x * 8 + (threadIdx.x >> 5);
  const int mt = gw >> 2;  // row tile (16 rows)
  const int nt = gw & 3;   // col tile (64/16)
  if (mt >= n_row_tiles) return;  // wave-uniform; EXEC all-1 for WMMA

  const int lane = threadIdx.x & 31;
  const int half = lane >> 4;  // 0: K={0,1}, 1: K={2,3}
  const int l    = lane & 15;
  const int kb   = half * 2;
  const int arow = mt * 16 + l;
  const int col  = nt * 16 + l;

  v8f acc = {};
#pragma unroll 4
  for (int k = 0; k < K; k += 4) {
    v2f a, b;
    a.x = A[(size_t)arow * lda + a_col_ofs + k + kb];
    a.y = A[(size_t)arow * lda + a_col_ofs + k + kb + 1];
    b.x = Wlds[(k + kb) * WSTRIDE + col];
    b.y = Wlds[(k + kb + 1) * WSTRIDE + col];
    acc = __builtin_amdgcn_wmma_f32_16x16x4_f32(false, a, false, b,
                                                (short)0, acc, false, false);
  }

  const float bc = bias[col];
#pragma unroll
  for (int g = 0; g < 8; ++g) {
    const int r = mt * 16 + half * 8 + g;  // D VGPR g -> row g / g+8
    float* yp = Y + (size_t)r * FDIM + y_col_ofs + col;
    float val;
    if (UPDATE) {
      val = *yp + acc[g] + deg[r] * bc;
    } else {
      val = acc[g] + bc;
    }
    *yp = fmaxf(val, 0.0f);
  }
}

// One wave per edge: gather 128-float row of X[src], atomic-add into P[dst].
__global__ void scatter_edges(const float* __restrict__ X,
                              const int* __restrict__ src,
                              const int* __restrict__ dst,
                              float* __restrict__ P, int E) {
  const int e = blockIdx.x * (blockDim.x >> 5) + (threadIdx.x >> 5);
  if (e >= E) return;
  const int lane = threadIdx.x & 31;
  const int s = src[e];
  const int d = dst[e];
  const float4 val = ((const float4*)(X + (size_t)s * FDIM))[lane];
  float* pd = P + (size_t)d * FDIM + lane * 4;
  atomicAdd(pd + 0, val.x);
  atomicAdd(pd + 1, val.y);
  atomicAdd(pd + 2, val.z);
  atomicAdd(pd + 3, val.w);
}

extern "C" void kernel_launch(void* const* d_in, const int* in_sizes, int n_in,
                              void* d_out, int out_size, void* d_ws, size_t ws_size,
                              hipStream_t stream) {
  const float* s   = (const float*)d_in[0];
  const float* v   = (const float*)d_in[1];
  const int*   eix = (const int*)d_in[2];  // edge_index [2,E] int32
  const float* Ws  = (const float*)d_in[3];
  const float* bs  = (const float*)d_in[4];
  const float* Wv  = (const float*)d_in[5];
  const float* bv  = (const float*)d_in[6];
  const float* Wms = (const float*)d_in[7];
  const float* bms = (const float*)d_in[8];
  const float* Wmv = (const float*)d_in[9];
  const float* bmv = (const float*)d_in[10];

  const int N = in_sizes[0] / 128;  // 50000 (divisible by 16)
  const int E = in_sizes[2] / 2;    // 800000
  const int* src = eix;
  const int* dst = eix + E;

  float* X   = (float*)d_out;         // [N,128] state == final output
  float* P   = (float*)d_ws;          // [N,128] edge aggregation
  float* deg = P + (size_t)N * FDIM;  // [N]

  const int row_tiles = N / 16;  // 3125
  const int wpb = 8;             // 8 waves per 256-thread block
  const int gemm_blocks = (row_tiles * 4 + wpb - 1) / wpb;
  const int scat_blocks = (E + wpb - 1) / wpb;

  // degree (recomputed each call: deterministic under graph replay)
  zero_f32v4<<<(N / 4 + 255) / 256, 256, 0, stream>>>((float4*)deg, N / 4);
  deg_count<<<(E + 255) / 256, 256, 0, stream>>>(dst, deg, E);

  // encode: X[:, :64] = relu(s@Ws+bs), X[:, 64:] = relu(v@Wv+bv)
  gemm_fused<128, false><<<gemm_blocks, 256, 0, stream>>>(
      s, 128, 0, Ws, bs, nullptr, X, 0, row_tiles);
  gemm_fused<128, false><<<gemm_blocks, 256, 0, stream>>>(
      v, 128, 0, Wv, bv, nullptr, X, HDIM, row_tiles);

  for (int it = 0; it < 2; ++it) {
    zero_f32v4<<<((N * FDIM / 4) + 255) / 256, 256, 0, stream>>>(
        (float4*)P, N * FDIM / 4);
    scatter_edges<<<scat_blocks, 256, 0, stream>>>(X, src, dst, P, E);
    // X = relu(X + segment_sum(X[src]) @ Wm + deg*b)   [linearity of segment_sum]
    gemm_fused<HDIM, true><<<gemm_blocks, 256, 0, stream>>>(
        P, FDIM, 0, Wms, bms, deg, X, 0, row_tiles);
    gemm_fused<HDIM, true><<<gemm_blocks, 256, 0, stream>>>(
        P, FDIM, HDIM, Wmv, bmv, deg, X, HDIM, row_tiles);
  }
}